// GINE_68367289418046
// MI455X (gfx1250) — compile-verified
//
#include <hip/hip_runtime.h>

// ---------------------------------------------------------------------------
// GINE (3-layer) + linear readout for MI455X (gfx1250).
// GEMMs: native fp32 WMMA (V_WMMA_F32_16X16X4_F32), 16x64 tile per wave32.
// Weights pre-swizzled into per-lane fragment order -> 1 b64 load per B frag.
// Edge aggregation: no-return global fp32 atomics (accumulator is L2-resident).
// ---------------------------------------------------------------------------

#define N_NODES 200000
#define N_EDGES 600000
#define DIM     128
#define DIM2    256
#define MTILES  (N_NODES / 16)          // 12500, exact
#define MBLOCKS ((MTILES + 7) / 8)      // 1563

typedef __attribute__((ext_vector_type(2))) float v2f;
typedef __attribute__((ext_vector_type(8))) float v8f;

__device__ __forceinline__ v8f wmma4(v2f a, v2f b, v8f c) {
  // D = A(16x4) * B(4x16) + C(16x16), fp32 in / fp32 accumulate
  return __builtin_amdgcn_wmma_f32_16x16x4_f32(
      /*neg_a=*/false, a, /*neg_b=*/false, b,
      /*c_mod=*/(short)0, c, /*reuse_a=*/false, /*reuse_b=*/false);
}

// A fragment (row-major A[M x K], lda = K):
// lanes 0-15: row mbase+lane, K = {kbase, kbase+1};
// lanes 16-31: row mbase+lane-16, K = {kbase+2, kbase+3}.
__device__ __forceinline__ v2f load_a(const float* __restrict__ A, int lda,
                                      int mbase, int kbase, int lane) {
  const size_t row = (size_t)(mbase + (lane & 15));
  const int kk = kbase + ((lane >> 4) << 1);
  return *(const v2f*)(A + row * lda + kk);
}

// ---------------------------------------------------------------------------
// One-time weight swizzle: W[K x Ncol] row-major -> fragment stream Ws where
// fragment (k4, n16) occupies 32 consecutive v2f (one per lane):
//   lane: col = n16*16 + (lane&15), rows kk = 4*k4 + 2*(lane>>4), kk+1.
// B fragment load in the GEMMs is then a single coalesced b64.
// ---------------------------------------------------------------------------
__global__ void __launch_bounds__(256) swizzle_w_kernel(
    const float* __restrict__ W, v2f* __restrict__ Ws, int K, int Ncol) {
  const int idx = blockIdx.x * 256 + threadIdx.x;
  const int NT = Ncol / 16;
  const int total = (K / 4) * NT * 32;
  if (idx >= total) return;
  const int lane = idx & 31;
  const int t = idx >> 5;
  const int n16 = t % NT;
  const int k4 = t / NT;
  const int n = n16 * 16 + (lane & 15);
  const int kk = 4 * k4 + ((lane >> 4) << 1);
  v2f b;
  b.x = W[(size_t)kk * Ncol + n];
  b.y = W[(size_t)(kk + 1) * Ncol + n];
  Ws[idx] = b;
}

// ---------------------------------------------------------------------------
// z = (1 + eps[l]) * h            (vectorized float4, N*D/4 threads)
// ---------------------------------------------------------------------------
__global__ void __launch_bounds__(256) init_z_kernel(
    float* __restrict__ Z, const float* __restrict__ H,
    const float* __restrict__ eps, int l) {
  const size_t i = (size_t)blockIdx.x * 256 + threadIdx.x;   // N*D/4 exact
  const float s = 1.0f + eps[l];
  float4 v = ((const float4*)H)[i];
  float4 o;
  o.x = s * v.x; o.y = s * v.y; o.z = s * v.z; o.w = s * v.w;
  ((float4*)Z)[i] = o;
}

// ---------------------------------------------------------------------------
// z[dst] += relu(h[src] + edge_attr)   (one thread per edge x float4)
// ---------------------------------------------------------------------------
__global__ void __launch_bounds__(256) scatter_kernel(
    float* __restrict__ Z, const float* __restrict__ H,
    const float* __restrict__ EA, const int* __restrict__ EI) {
  const size_t idx = (size_t)blockIdx.x * 256 + threadIdx.x;  // E*32 exact
  const int e  = (int)(idx >> 5);
  const int d4 = (int)(idx & 31);
  const int src = EI[e];
  const int dst = EI[N_EDGES + e];
  float4 hv = ((const float4*)(H  + (size_t)src * DIM))[d4];
  float4 ev = ((const float4*)(EA + (size_t)e   * DIM))[d4];
  float m0 = fmaxf(hv.x + ev.x, 0.0f);
  float m1 = fmaxf(hv.y + ev.y, 0.0f);
  float m2 = fmaxf(hv.z + ev.z, 0.0f);
  float m3 = fmaxf(hv.w + ev.w, 0.0f);
  float* zp = Z + (size_t)dst * DIM + d4 * 4;
  unsafeAtomicAdd(zp + 0, m0);
  unsafeAtomicAdd(zp + 1, m1);
  unsafeAtomicAdd(zp + 2, m2);
  unsafeAtomicAdd(zp + 3, m3);
}

// ---------------------------------------------------------------------------
__global__ void zero_stats_kernel(float* __restrict__ stats) {
  stats[threadIdx.x] = 0.0f;   // 512 threads: bnsum[256] + bnsq[256]
}

// ---------------------------------------------------------------------------
// U = Z @ W1 + b1  (K=128, Ncol=256), plus per-column sum / sumsq for BN.
// Wave tile: 16 rows x 64 cols (4 accumulators). Block: 8 waves = 128 rows.
// grid.y = 4 column groups of 64.
// ---------------------------------------------------------------------------
__global__ void __launch_bounds__(256) gemm1_kernel(
    const float* __restrict__ Z, const v2f* __restrict__ W1s,
    const float* __restrict__ b1, float* __restrict__ U,
    float* __restrict__ bnsum, float* __restrict__ bnsq) {
  __shared__ float lds_s[4][8][32];
  __shared__ float lds_q[4][8][32];
  const int lane  = threadIdx.x & 31;
  const int wave  = threadIdx.x >> 5;
  const int mtile = blockIdx.x * 8 + wave;
  const int nbase = blockIdx.y * 64;
  const int n16   = blockIdx.y * 4;
  const int mbase = mtile * 16;
  const bool active = (mtile < MTILES);   // wave-uniform; EXEC all-1 in WMMA
  const int NT = DIM2 / 16;               // 16 fragment columns

  v8f acc[4] = {};
  if (active) {
    for (int k4 = 0; k4 < DIM / 4; ++k4) {
      v2f a = load_a(Z, DIM, mbase, 4 * k4, lane);
      #pragma unroll
      for (int g = 0; g < 4; ++g) {
        v2f b = W1s[((size_t)k4 * NT + n16 + g) * 32 + lane];
        acc[g] = wmma4(a, b, acc[g]);
      }
    }
  }
  // epilogue: u = acc + bias; store + BN partial stats
  const int cl    = lane & 15;
  const int mrow0 = mbase + ((lane >> 4) << 3);
  #pragma unroll
  for (int g = 0; g < 4; ++g) {
    const int col = nbase + g * 16 + cl;
    const float bias = b1[col];
    float s = 0.0f, q = 0.0f;
    if (active) {
      #pragma unroll
      for (int r = 0; r < 8; ++r) {
        float u = acc[g][r] + bias;
        U[(size_t)(mrow0 + r) * DIM2 + col] = u;
        s += u;
        q += u * u;
      }
    }
    lds_s[g][wave][lane] = s;
    lds_q[g][wave][lane] = q;
  }
  __syncthreads();
  if (threadIdx.x < 64) {
    const int g = threadIdx.x >> 4;
    const int c = threadIdx.x & 15;
    float ts = 0.0f, tq = 0.0f;
    #pragma unroll
    for (int w = 0; w < 8; ++w) {
      ts += lds_s[g][w][c] + lds_s[g][w][c + 16];
      tq += lds_q[g][w][c] + lds_q[g][w][c + 16];
    }
    unsafeAtomicAdd(&bnsum[nbase + g * 16 + c], ts);
    unsafeAtomicAdd(&bnsq [nbase + g * 16 + c], tq);
  }
}

// ---------------------------------------------------------------------------
// Fold BN into per-column scale/shift: y = scale*u + shift
// ---------------------------------------------------------------------------
__global__ void bnfinal_kernel(
    const float* __restrict__ bnsum, const float* __restrict__ bnsq,
    const float* __restrict__ gamma, const float* __restrict__ beta,
    float* __restrict__ scale, float* __restrict__ shift, int l) {
  const int c = threadIdx.x;                 // 256 threads
  const float inv_n = 1.0f / (float)N_NODES;
  const float mu  = bnsum[c] * inv_n;
  const float var = bnsq[c] * inv_n - mu * mu;
  const float iv  = __frsqrt_rn(var + 1e-5f);
  const float sc  = gamma[l * DIM2 + c] * iv;
  scale[c] = sc;
  shift[c] = beta[l * DIM2 + c] - mu * sc;
}

// ---------------------------------------------------------------------------
// H = relu( relu(scale*U + shift) @ W2 + b2 )   (K=256, Ncol=128)
// BN + inner ReLU fused into the A-fragment load. grid.y = 2 (64-col groups).
// ---------------------------------------------------------------------------
__global__ void __launch_bounds__(256) gemm2_kernel(
    const float* __restrict__ U, const v2f* __restrict__ W2s,
    const float* __restrict__ b2, const float* __restrict__ scale,
    const float* __restrict__ shift, float* __restrict__ Hout) {
  const int lane  = threadIdx.x & 31;
  const int wave  = threadIdx.x >> 5;
  const int mtile = blockIdx.x * 8 + wave;
  if (mtile >= MTILES) return;             // wave-uniform exit
  const int nbase = blockIdx.y * 64;
  const int n16   = blockIdx.y * 4;
  const int mbase = mtile * 16;
  const int NT = DIM / 16;                 // 8 fragment columns

  v8f acc[4] = {};
  for (int k4 = 0; k4 < DIM2 / 4; ++k4) {
    const int kk = 4 * k4 + ((lane >> 4) << 1);
    v2f sc = *(const v2f*)(scale + kk);
    v2f sh = *(const v2f*)(shift + kk);
    v2f u  = load_a(U, DIM2, mbase, 4 * k4, lane);
    v2f a;
    a.x = fmaxf(fmaf(sc.x, u.x, sh.x), 0.0f);
    a.y = fmaxf(fmaf(sc.y, u.y, sh.y), 0.0f);
    #pragma unroll
    for (int g = 0; g < 4; ++g) {
      v2f b = W2s[((size_t)k4 * NT + n16 + g) * 32 + lane];
      acc[g] = wmma4(a, b, acc[g]);
    }
  }
  const int cl    = lane & 15;
  const int mrow0 = mbase + ((lane >> 4) << 3);
  #pragma unroll
  for (int g = 0; g < 4; ++g) {
    const int col = nbase + g * 16 + cl;
    const float bias = b2[col];
    #pragma unroll
    for (int r = 0; r < 8; ++r)
      Hout[(size_t)(mrow0 + r) * DIM + col] = fmaxf(acc[g][r] + bias, 0.0f);
  }
}

// ---------------------------------------------------------------------------
// out = concat(h0,h1,h2,h3) @ lin_w + lin_b   (K=512 as 4x128 segments)
// ---------------------------------------------------------------------------
__global__ void __launch_bounds__(256) gemm3_kernel(
    const float* __restrict__ h0, const float* __restrict__ h1,
    const float* __restrict__ h2, const float* __restrict__ h3,
    const v2f* __restrict__ LWs, const float* __restrict__ lb,
    float* __restrict__ Out) {
  const int lane  = threadIdx.x & 31;
  const int wave  = threadIdx.x >> 5;
  const int mtile = blockIdx.x * 8 + wave;
  if (mtile >= MTILES) return;             // wave-uniform exit
  const int nbase = blockIdx.y * 64;
  const int n16   = blockIdx.y * 4;
  const int mbase = mtile * 16;
  const int NT = DIM / 16;                 // 8 fragment columns

  const float* seg[4] = {h0, h1, h2, h3};
  v8f acc[4] = {};
  #pragma unroll
  for (int sgi = 0; sgi < 4; ++sgi) {
    const float* A = seg[sgi];
    for (int k4 = 0; k4 < DIM / 4; ++k4) {
      v2f a = load_a(A, DIM, mbase, 4 * k4, lane);
      const size_t kf = (size_t)(sgi * (DIM / 4) + k4) * NT;
      #pragma unroll
      for (int g = 0; g < 4; ++g) {
        v2f b = LWs[(kf + n16 + g) * 32 + lane];
        acc[g] = wmma4(a, b, acc[g]);
      }
    }
  }
  const int cl    = lane & 15;
  const int mrow0 = mbase + ((lane >> 4) << 3);
  #pragma unroll
  for (int g = 0; g < 4; ++g) {
    const int col = nbase + g * 16 + cl;
    const float bias = lb[col];
    #pragma unroll
    for (int r = 0; r < 8; ++r)
      Out[(size_t)(mrow0 + r) * DIM + col] = acc[g][r] + bias;
  }
}

// ---------------------------------------------------------------------------
extern "C" void kernel_launch(void* const* d_in, const int* in_sizes, int n_in,
                              void* d_out, int out_size, void* d_ws, size_t ws_size,
                              hipStream_t stream) {
  const float* x     = (const float*)d_in[0];
  const int*   ei    = (const int*)  d_in[1];
  const float* ea    = (const float*)d_in[2];
  const float* eps   = (const float*)d_in[3];
  const float* W1    = (const float*)d_in[4];
  const float* b1    = (const float*)d_in[5];
  const float* gamma = (const float*)d_in[6];
  const float* beta  = (const float*)d_in[7];
  const float* W2    = (const float*)d_in[8];
  const float* b2    = (const float*)d_in[9];
  const float* lw    = (const float*)d_in[10];
  const float* lb    = (const float*)d_in[11];
  float* out = (float*)d_out;

  float* ws = (float*)d_ws;
  const size_t ND = (size_t)N_NODES * DIM;
  float* Z     = ws;                 // N*128
  float* U     = ws + ND;            // N*256
  float* H1    = ws + 3 * ND;        // N*128
  float* H2    = ws + 4 * ND;
  float* H3    = ws + 5 * ND;
  float* stats = ws + 6 * ND;        // bnsum[256] | bnsq[256] | scale[256] | shift[256]
  float* wswz  = ws + 6 * ND + 1024; // swizzled weights
  const size_t W1SZ = (size_t)DIM * DIM2;        // 32768 floats per layer
  const size_t W2SZ = (size_t)DIM2 * DIM;        // 32768 floats per layer
  v2f* W1s[3] = {(v2f*)(wswz),              (v2f*)(wswz + W1SZ),     (v2f*)(wswz + 2 * W1SZ)};
  v2f* W2s[3] = {(v2f*)(wswz + 3 * W1SZ),   (v2f*)(wswz + 3 * W1SZ + W2SZ),
                 (v2f*)(wswz + 3 * W1SZ + 2 * W2SZ)};
  v2f* LWs    = (v2f*)(wswz + 3 * W1SZ + 3 * W2SZ);

  // One-time weight swizzles (fragment-order, contiguous per-lane v2f).
  for (int l = 0; l < 3; ++l) {
    swizzle_w_kernel<<<(DIM  / 4) * (DIM2 / 16) * 32 / 256, 256, 0, stream>>>(
        W1 + (size_t)l * W1SZ, W1s[l], DIM, DIM2);
    swizzle_w_kernel<<<(DIM2 / 4) * (DIM  / 16) * 32 / 256, 256, 0, stream>>>(
        W2 + (size_t)l * W2SZ, W2s[l], DIM2, DIM);
  }
  swizzle_w_kernel<<<(4 * DIM / 4) * (DIM / 16) * 32 / 256, 256, 0, stream>>>(
      lw, LWs, 4 * DIM, DIM);

  const float* Hin = x;
  float* Houts[3] = {H1, H2, H3};

  const int gridZ = (int)(ND / 4 / 256);          // 25000, exact
  const int gridE = N_EDGES * (DIM / 4) / 256;    // 75000, exact

  for (int l = 0; l < 3; ++l) {
    zero_stats_kernel<<<1, 512, 0, stream>>>(stats);
    init_z_kernel<<<gridZ, 256, 0, stream>>>(Z, Hin, eps, l);
    scatter_kernel<<<gridE, 256, 0, stream>>>(Z, Hin, ea, ei);
    gemm1_kernel<<<dim3(MBLOCKS, 4), 256, 0, stream>>>(
        Z, W1s[l], b1 + l * DIM2, U, stats, stats + 256);
    bnfinal_kernel<<<1, 256, 0, stream>>>(
        stats, stats + 256, gamma, beta, stats + 512, stats + 768, l);
    gemm2_kernel<<<dim3(MBLOCKS, 2), 256, 0, stream>>>(
        U, W2s[l], b2 + l * DIM, stats + 512, stats + 768, Houts[l]);
    Hin = Houts[l];
  }
  gemm3_kernel<<<dim3(MBLOCKS, 2), 256, 0, stream>>>(
      x, H1, H2, H3, LWs, lb, out);
}